// Deform_86294482911263
// MI455X (gfx1250) — compile-verified
//
#include <hip/hip_runtime.h>
#include <hip/hip_bf16.h>

typedef __attribute__((ext_vector_type(2))) float v2f;
typedef __attribute__((ext_vector_type(8))) float v8f;
typedef __attribute__((ext_vector_type(2))) int   v2i;

#define AS1 __attribute__((address_space(1)))
#define AS3 __attribute__((address_space(3)))

#define GRID_N   256
#define PAD      30
#define GLOBAL_N 316            // GRID + 2*PAD
#define INP      512
#define NB       16             // batch
#define KS       61             // gaussian kernel size
#define NF       48             // 16 batches * 3 conv fields
#define FSTRIDE  320            // padded row stride for fields

// workspace layout (in floats)
#define OFF_F     0
#define SZ_F      ((size_t)NF * GLOBAL_N * FSTRIDE)       // ~19.4 MB
#define OFF_CONV  (OFF_F + SZ_F)
#define SZ_CONV   ((size_t)NF * GRID_N * GRID_N)          // ~12.6 MB
#define OFF_GRIDS (OFF_CONV + SZ_CONV)
#define SZ_GRIDS  ((size_t)NB * 2 * GRID_N * GRID_N)      // ~8.4 MB

#if __has_builtin(__builtin_amdgcn_global_load_async_to_lds_b64)
#define HAVE_ASYNC_LDS 1
#endif

__device__ __forceinline__ int clampi(int v, int lo, int hi) {
    return v < lo ? lo : (v > hi ? hi : v);
}

__device__ __forceinline__ void wait_async0() {
#if __has_builtin(__builtin_amdgcn_s_wait_asynccnt)
    __builtin_amdgcn_s_wait_asynccnt(0);
#else
    asm volatile("s_wait_asynccnt 0x0" ::: "memory");
#endif
}

// ---------------------------------------------------------------------------
// Stage 1: build the three conv input fields per batch:
//   F[b][0] = edge-padded xs ; F[b][1] = lin[x]*xs_hm ; F[b][2] = lin[y]*xs_hm
//   lin[i] = (i - PAD) / (GRID_N - 1)
// ---------------------------------------------------------------------------
__global__ void prep_fields(const float* __restrict__ xs, float* __restrict__ F) {
    int idx = blockIdx.x * blockDim.x + threadIdx.x;
    int total = NF * GLOBAL_N * GLOBAL_N;
    if (idx >= total) return;
    int x = idx % GLOBAL_N;
    int t = idx / GLOBAL_N;
    int y = t % GLOBAL_N;
    int f = t / GLOBAL_N;
    int b = f / 3, c = f % 3;
    int sy = clampi(y - PAD, 0, GRID_N - 1);
    int sx = clampi(x - PAD, 0, GRID_N - 1);
    float v = xs[((size_t)b * GRID_N + sy) * GRID_N + sx];
    float w = 1.0f;
    if (c == 1)      w = (float)(x - PAD) * (1.0f / (GRID_N - 1));
    else if (c == 2) w = (float)(y - PAD) * (1.0f / (GRID_N - 1));
    F[((size_t)f * GLOBAL_N + y) * FSTRIDE + x] = v * w;
}

// ---------------------------------------------------------------------------
// Stage 2: 61x61 VALID correlation via V_WMMA_F32_16X16X4_F32.
// Block = 128 threads (4 waves), output region 32 rows x 64 cols.
// Each wave owns one 16-col strip and accumulates TWO stacked 16x16 tiles
// (rows y0 and y0+16) so the banded-gauss B fragment is loaded once per two
// WMMAs.  Per ky: C[t] += A[t][16x80] x B[80x16], A = LDS input rows (b64
// aligned loads, stride 130), B = banded gauss row (gpad[ky][k-n+16]).
// Tile fill uses gfx1250 async global->LDS (ASYNCcnt) when available.
// ---------------------------------------------------------------------------
#define TROWS 92          // 32 + 60 input rows
#define TCOLS 124         // 64 + 60 valid input cols
#define TSTR  130         // even stride: 8B-aligned b64 A loads, banks ok
#define GPSTR 96          // banded gauss row stride: j+16 in [0,96)

__global__ void __launch_bounds__(128) conv_wmma(const float* __restrict__ F,
                                                 const float* __restrict__ gauss,
                                                 float* __restrict__ out) {
    __shared__ float tile[TROWS * TSTR];   // 47.8 KB
    __shared__ float gpad[KS * GPSTR];     // 23.4 KB   (total 71.3 KB / WG)

    int tid = threadIdx.x;
    int bid = blockIdx.x;
    int f   = bid >> 5;        // field 0..47
    int rem = bid & 31;
    int ty  = rem >> 2;        // y block 0..7 (32 rows each)
    int xg  = rem & 3;         // x group 0..3 (64 cols each)
    int y0  = ty * 32;
    int X0  = xg * 64;

    // banded gauss: gpad[ky][j+16] = gauss[ky][j] for j in [0,61), else 0
    for (int i = tid; i < KS * GPSTR; i += 128) {
        int ky = i / GPSTR;
        int j  = (i % GPSTR) - 16;
        gpad[i] = (j >= 0 && j < KS) ? gauss[ky * KS + j] : 0.0f;
    }

    const float* Fb = F + (size_t)f * GLOBAL_N * FSTRIDE;
#ifdef HAVE_ASYNC_LDS
    // zero the tail columns so out-of-band taps hit real zeros
    for (int i = tid; i < TROWS * (TSTR - TCOLS); i += 128) {
        int r = i / (TSTR - TCOLS);
        int c = TCOLS + i % (TSTR - TCOLS);
        tile[r * TSTR + c] = 0.0f;
    }
    // async copy the valid 92x124 region as b64 (both sides 8B aligned)
    for (int i = tid; i < TROWS * (TCOLS / 2); i += 128) {
        int r  = i / (TCOLS / 2);
        int c2 = (i % (TCOLS / 2)) * 2;
        const float* gsrc = Fb + (size_t)(y0 + r) * FSTRIDE + X0 + c2;
        float*       ldst = &tile[r * TSTR + c2];
        __builtin_amdgcn_global_load_async_to_lds_b64(
            (AS1 v2i*)(unsigned long long)(const void*)gsrc,
            (AS3 v2i*)(void*)ldst, 0, 0);
    }
    wait_async0();
#else
    for (int i = tid; i < TROWS * TSTR; i += 128) {
        int r = i / TSTR;
        int c = i % TSTR;
        tile[i] = (c < TCOLS) ? Fb[(size_t)(y0 + r) * FSTRIDE + X0 + c] : 0.0f;
    }
#endif
    __syncthreads();

    int lane  = tid & 31;
    int wave  = tid >> 5;
    int half  = lane >> 4;       // 0: lanes 0-15, 1: lanes 16-31
    int n     = lane & 15;       // == m (A row) == n (B/C col) for this lane
    int xbase = wave * 16;

    v8f c0 = {0.f, 0.f, 0.f, 0.f, 0.f, 0.f, 0.f, 0.f};
    v8f c1 = {0.f, 0.f, 0.f, 0.f, 0.f, 0.f, 0.f, 0.f};
    const int jb = 2 * half - n + 16;   // gpad col for s=0 (k = 2*half)

    for (int ky = 0; ky < KS; ++ky) {
        const float* gr  = &gpad[ky * GPSTR];
        const float* ir0 = &tile[(n + ky) * TSTR + xbase];  // A row m = lane&15
        const float* ir1 = ir0 + 16 * TSTR;                 // second M tile
        #pragma unroll 4
        for (int s = 0; s < 20; ++s) {
            int kk = 4 * s + 2 * half;
            v2f a0 = *(const v2f*)(ir0 + kk);   // ds_load_b64 (8B aligned)
            v2f a1 = *(const v2f*)(ir1 + kk);
            v2f bb;
            bb.x = gr[jb + 4 * s];              // B[kk][n]   = gauss[ky][kk-n]
            bb.y = gr[jb + 4 * s + 1];          // B[kk+1][n]
            c0 = __builtin_amdgcn_wmma_f32_16x16x4_f32(
                     false, a0, false, bb, (short)0, c0, false, false);
            c1 = __builtin_amdgcn_wmma_f32_16x16x4_f32(
                     false, a1, false, bb, (short)0, c1, false, false);
        }
    }

    // C layout: VGPR r -> row (tile base) + r + 8*half, col X0 + xbase + n
    float* ob = out + (size_t)f * GRID_N * GRID_N;
    #pragma unroll
    for (int r = 0; r < 8; ++r) {
        ob[(size_t)(y0 + r + 8 * half) * GRID_N + X0 + xbase + n] = c0[r];
    }
    #pragma unroll
    for (int r = 0; r < 8; ++r) {
        ob[(size_t)(y0 + 16 + r + 8 * half) * GRID_N + X0 + xbase + n] = c1[r];
    }
}

// ---------------------------------------------------------------------------
// Stage 3: grids = clip(all_filter / p_filter * 2 - 1, -1, 1)
// ---------------------------------------------------------------------------
__global__ void make_grids(const float* __restrict__ conv, float* __restrict__ grids) {
    int idx = blockIdx.x * blockDim.x + threadIdx.x;
    int total = NB * GRID_N * GRID_N;
    if (idx >= total) return;
    int x = idx % GRID_N;
    int t = idx / GRID_N;
    int y = t % GRID_N;
    int b = t / GRID_N;
    const size_t plane = (size_t)GRID_N * GRID_N;
    const float* base = conv + (size_t)b * 3 * plane;
    float pf = base[(size_t)y * GRID_N + x];
    float a0 = base[plane + (size_t)y * GRID_N + x];
    float a1 = base[2 * plane + (size_t)y * GRID_N + x];
    float g0 = fminf(fmaxf(a0 / pf * 2.0f - 1.0f, -1.0f), 1.0f);
    float g1 = fminf(fmaxf(a1 / pf * 2.0f - 1.0f, -1.0f), 1.0f);
    grids[((size_t)(b * 2 + 0) * GRID_N + y) * GRID_N + x] = g0;
    grids[((size_t)(b * 2 + 1) * GRID_N + y) * GRID_N + x] = g1;
}

// ---------------------------------------------------------------------------
// Stage 4: jax.image.resize bilinear 256->512 (half-pixel centers; boundary
// weight renormalization == clamp-to-edge for the linear kernel).
// ---------------------------------------------------------------------------
__global__ void resize_bilinear(const float* __restrict__ grids, float* __restrict__ grid) {
    int idx = blockIdx.x * blockDim.x + threadIdx.x;
    int total = NB * 2 * INP * INP;
    if (idx >= total) return;
    int x = idx % INP;
    int t = idx / INP;
    int y = t % INP;
    int bc = t / INP;   // b*2 + ch

    float sy = fminf(fmaxf(0.5f * y - 0.25f, 0.0f), (float)(GRID_N - 1));
    float sx = fminf(fmaxf(0.5f * x - 0.25f, 0.0f), (float)(GRID_N - 1));
    int y0i = (int)floorf(sy);
    int x0i = (int)floorf(sx);
    int y1i = min(y0i + 1, GRID_N - 1);
    int x1i = min(x0i + 1, GRID_N - 1);
    float tyw = sy - (float)y0i;
    float txw = sx - (float)x0i;

    const float* gp = grids + (size_t)bc * GRID_N * GRID_N;
    float v = (1.0f - tyw) * ((1.0f - txw) * gp[(size_t)y0i * GRID_N + x0i] +
                              txw         * gp[(size_t)y0i * GRID_N + x1i]) +
              tyw         * ((1.0f - txw) * gp[(size_t)y1i * GRID_N + x0i] +
                              txw         * gp[(size_t)y1i * GRID_N + x1i]);
    grid[idx] = v;   // [B,2,512,512] flat == output layout
}

// ---------------------------------------------------------------------------
// Stage 5: grid_sample bilinear, align_corners=True, zeros padding.
// grid_r = (49*identity + grid_l) / 50
// ---------------------------------------------------------------------------
__global__ void sample_kernel(const float* __restrict__ x,
                              const float* __restrict__ grid,
                              float* __restrict__ out) {
    int idx = blockIdx.x * blockDim.x + threadIdx.x;
    int total = NB * INP * INP;
    if (idx >= total) return;
    int px = idx % INP;
    int t  = idx / INP;
    int py = t % INP;
    int b  = t / INP;

    const float inv511 = 1.0f / 511.0f;
    float lix = -1.0f + 2.0f * (float)px * inv511;
    float liy = -1.0f + 2.0f * (float)py * inv511;

    const size_t plane = (size_t)INP * INP;
    const float* gb = grid + (size_t)b * 2 * plane;
    float gl0 = gb[(size_t)py * INP + px];
    float gl1 = gb[plane + (size_t)py * INP + px];
    float grx = (49.0f * lix + gl0) * (1.0f / 50.0f);
    float gry = (49.0f * liy + gl1) * (1.0f / 50.0f);

    float xc = (grx + 1.0f) * 0.5f * 511.0f;
    float yc = (gry + 1.0f) * 0.5f * 511.0f;
    float x0f = floorf(xc), y0f = floorf(yc);
    float wx1 = xc - x0f, wx0 = 1.0f - wx1;
    float wy1 = yc - y0f, wy0 = 1.0f - wy1;
    int x0i = (int)x0f, y0i = (int)y0f;
    int x1i = x0i + 1,  y1i = y0i + 1;
    bool vx0 = (x0i >= 0) && (x0i <= INP - 1);
    bool vx1 = (x1i >= 0) && (x1i <= INP - 1);
    bool vy0 = (y0i >= 0) && (y0i <= INP - 1);
    bool vy1 = (y1i >= 0) && (y1i <= INP - 1);
    int cx0 = clampi(x0i, 0, INP - 1), cx1 = clampi(x1i, 0, INP - 1);
    int cy0 = clampi(y0i, 0, INP - 1), cy1 = clampi(y1i, 0, INP - 1);

    #pragma unroll
    for (int c = 0; c < 3; ++c) {
        const float* xb = x + ((size_t)(b * 3 + c)) * plane;
        float v00 = (vx0 && vy0) ? xb[(size_t)cy0 * INP + cx0] : 0.0f;
        float v10 = (vx1 && vy0) ? xb[(size_t)cy0 * INP + cx1] : 0.0f;
        float v01 = (vx0 && vy1) ? xb[(size_t)cy1 * INP + cx0] : 0.0f;
        float v11 = (vx1 && vy1) ? xb[(size_t)cy1 * INP + cx1] : 0.0f;
        out[((size_t)(b * 3 + c)) * plane + (size_t)py * INP + px] =
            v00 * (wx0 * wy0) + v10 * (wx1 * wy0) +
            v01 * (wx0 * wy1) + v11 * (wx1 * wy1);
    }
}

// ---------------------------------------------------------------------------
extern "C" void kernel_launch(void* const* d_in, const int* in_sizes, int n_in,
                              void* d_out, int out_size, void* d_ws, size_t ws_size,
                              hipStream_t stream) {
    const float* x     = (const float*)d_in[0];   // [16,3,512,512]
    const float* xs    = (const float*)d_in[1];   // [16,1,256,256]
    const float* gauss = (const float*)d_in[2];   // [61,61]
    float* out = (float*)d_out;                   // x_sampled (16*3*512*512) ++ grid (16*2*512*512)
    float* ws  = (float*)d_ws;                    // needs ~40.4 MB

    float* F     = ws + OFF_F;
    float* conv  = ws + OFF_CONV;
    float* grids = ws + OFF_GRIDS;
    float* grid_out = out + (size_t)NB * 3 * INP * INP;

    {
        int total = NF * GLOBAL_N * GLOBAL_N;
        prep_fields<<<(total + 255) / 256, 256, 0, stream>>>(xs, F);
    }
    conv_wmma<<<NF * 8 * 4, 128, 0, stream>>>(F, gauss, conv);
    {
        int total = NB * GRID_N * GRID_N;
        make_grids<<<(total + 255) / 256, 256, 0, stream>>>(conv, grids);
    }
    {
        int total = NB * 2 * INP * INP;
        resize_bilinear<<<(total + 255) / 256, 256, 0, stream>>>(grids, grid_out);
    }
    {
        int total = NB * INP * INP;
        sample_kernel<<<(total + 255) / 256, 256, 0, stream>>>(x, grid_out, out);
    }
}